// SOMNetwork_81020263072064
// MI455X (gfx1250) — compile-verified
//
#include <hip/hip_runtime.h>
#include <math.h>

typedef float v2f __attribute__((ext_vector_type(2)));
typedef float v8f __attribute__((ext_vector_type(8)));

#define RBF_EPS 1e-12f

// ---- CDNA5 async copy: global -> LDS, 16B per lane, tracked by ASYNCcnt ----
__device__ __forceinline__ void async_g2lds_b128(uint32_t lds_byte_off,
                                                 const float* g) {
    asm volatile("global_load_async_to_lds_b128 %0, %1, off"
                 :: "v"(lds_byte_off), "v"((unsigned long long)(uintptr_t)g)
                 : "memory");
}
__device__ __forceinline__ void wait_async0() {
    asm volatile("s_wait_asynccnt 0x0" ::: "memory");
}

// ---------------- im2col for stage 1 (5x5 windows, stride 4, 28x28 -> 6x6) ----
__global__ void im2col1_k(const float* __restrict__ x, int c,
                          float* __restrict__ P, float* __restrict__ pn, int N) {
    int r = blockIdx.x * blockDim.x + threadIdx.x;
    if (r >= N * 36) return;
    int n = r / 36, q = r % 36, i = q / 6, j = q % 6;
    const float* src = x + (((size_t)n * 3 + c) * 28 + i * 4) * 28 + j * 4;
    float* dst = P + (size_t)r * 32;
    float norm = 0.f;
    #pragma unroll
    for (int kr = 0; kr < 5; kr++)
        #pragma unroll
        for (int kc = 0; kc < 5; kc++) {
            float v = src[kr * 28 + kc];
            dst[kr * 5 + kc] = v;
            norm += v * v;
        }
    #pragma unroll
    for (int k = 25; k < 32; k++) dst[k] = 0.f;
    pn[r] = norm;
}

// ---------------- weight prep: norms + WMMA-swizzled B (K x O, float2/lane) ---
__global__ void wprep_k(const float* __restrict__ w, float* __restrict__ bsw,
                        float* __restrict__ wn, int K, int Kpad, int O, int Opad) {
    int o = blockIdx.x * blockDim.x + threadIdx.x;
    if (o >= Opad) return;
    const float* wrow = w + (size_t)o * K;
    float norm = 0.f;
    if (o < O) for (int k = 0; k < K; k++) norm += wrow[k] * wrow[k];
    wn[o] = norm;
    v2f* b2 = (v2f*)bsw;
    int nt4 = Kpad >> 2;
    for (int t = 0; t < nt4; t++) {
        float v[4];
        #pragma unroll
        for (int u = 0; u < 4; u++) {
            int k = 4 * t + u;
            v[u] = (o < O && k < K) ? wrow[k] : 0.f;
        }
        b2[((size_t)t * Opad + o) * 2 + 0] = (v2f){v[0], v[1]};
        b2[((size_t)t * Opad + o) * 2 + 1] = (v2f){v[2], v[3]};
    }
}

// ---------------- distance GEMM via V_WMMA_F32_16X16X4_F32 ---------------------
// Block stages a (16*MT x Kpad) A panel in LDS via async-to-LDS copies; each
// wave owns MT 16x16 accumulators so one B fetch feeds MT WMMAs.
template <int MT>
__global__ void __launch_bounds__(256)
gemm_dist_k(const float* __restrict__ P, const float* __restrict__ Bsw,
            const float* __restrict__ pn, const float* __restrict__ wn,
            float* __restrict__ D, int Kpad, int O, int Opad) {
    extern __shared__ float ldsA[];
    const int rowBase = blockIdx.x * (16 * MT);

    // async stage A panel: 16 bytes per lane per issue
    {
        const int elems = 16 * MT * Kpad;                 // floats, mult of 4
        const uint32_t ldsBase = (uint32_t)(uintptr_t)ldsA; // low 32 bits give the LDS offset
        const float* gsrc = P + (size_t)rowBase * Kpad;
        for (int idx = threadIdx.x * 4; idx < elems; idx += blockDim.x * 4)
            async_g2lds_b128(ldsBase + (uint32_t)(idx * 4), gsrc + idx);
        wait_async0();
    }
    __syncthreads();

    const int lane = threadIdx.x & 31;     // wave32
    const int wave = threadIdx.x >> 5;
    const int half = lane >> 4;            // 0: lanes 0-15, 1: lanes 16-31
    const int l16  = lane & 15;
    const int koff = half * 2;             // A/B K sublane offset per ISA layout
    const int ntiles = Opad >> 4;
    const v2f* B2 = (const v2f*)Bsw;
    const float* arow = ldsA + l16 * Kpad + koff;

    for (int nt = wave; nt < ntiles; nt += (blockDim.x >> 5)) {
        int col = nt * 16 + l16;
        __builtin_prefetch(&Bsw[((size_t)col) * 4], 0, 1);
        v8f acc[MT];
        #pragma unroll
        for (int mt = 0; mt < MT; mt++) acc[mt] = (v8f){};
        for (int k0 = 0; k0 < Kpad; k0 += 4) {
            v2f b = B2[((size_t)(k0 >> 2) * Opad + col) * 2 + half];
            #pragma unroll
            for (int mt = 0; mt < MT; mt++) {
                v2f a = *(const v2f*)(arow + mt * 16 * Kpad + k0);
                acc[mt] = __builtin_amdgcn_wmma_f32_16x16x4_f32(
                    false, a, false, b, (short)0, acc[mt], false, false);
            }
        }
        if (col < O) {
            float wnorm = wn[col];
            #pragma unroll
            for (int mt = 0; mt < MT; mt++) {
                #pragma unroll
                for (int r = 0; r < 8; r++) {
                    int row = rowBase + mt * 16 + r + half * 8;
                    float d2 = pn[row] + wnorm - 2.f * acc[mt][r];
                    D[(size_t)row * O + col] = sqrtf(fmaxf(d2, 0.f) + RBF_EPS);
                }
            }
        }
    }
}

// ---------------- global std reduction: per-row partials + float atomics ------
__global__ void __launch_bounds__(128)
reduce_k(const float* __restrict__ D, float* sum, float* sumsq, int O, int posMod) {
    __shared__ float s1[128], s2[128];
    int r = blockIdx.x;
    const float* row = D + (size_t)r * O;
    float a = 0.f, b = 0.f;
    for (int k = threadIdx.x; k < O; k += 128) { float v = row[k]; a += v; b += v * v; }
    s1[threadIdx.x] = a; s2[threadIdx.x] = b; __syncthreads();
    for (int s = 64; s > 0; s >>= 1) {
        if (threadIdx.x < s) { s1[threadIdx.x] += s1[threadIdx.x + s];
                               s2[threadIdx.x] += s2[threadIdx.x + s]; }
        __syncthreads();
    }
    if (threadIdx.x == 0) {
        int p = (posMod > 1) ? (r % posMod) : 0;
        atomicAdd(&sum[p], s1[0]);
        atomicAdd(&sumsq[p], s2[0]);
    }
}

__global__ void zero_k(float* p, int n) {
    int i = blockIdx.x * blockDim.x + threadIdx.x;
    if (i < n) p[i] = 0.f;
}

__global__ void coef_k(const float* sum, const float* sumsq, float* coef,
                       int npos, float G) {
    int p = blockIdx.x * blockDim.x + threadIdx.x;
    if (p >= npos) return;
    float s = sum[p], q = sumsq[p];
    float var = (q - s * s / G) / (G - 1.0f);   // ddof=1, matches torch.std
    coef[p] = -0.5f / var;
}

// ---------------- finalize stage1: gauss+crelu + 2x2 alpha-pool -> P2 ---------
__global__ void f1_k(const float* __restrict__ D, const float* __restrict__ coef,
                     const float* __restrict__ cb, float* __restrict__ P2, int N) {
    int t = blockIdx.x * blockDim.x + threadIdx.x;
    int total = N * 9 * 112;
    if (t >= total) return;
    int k = t % 112, row2 = t / 112;
    float out = 0.f;
    if (k < 100) {
        int n = row2 / 9, q = row2 % 9, pi = q / 3, pj = q % 3;
        float bias = cb[0];
        const float al[4] = {0.729f, 0.81f, 0.9f, 1.0f};  // 0.9^[[3,2],[1,0]]
        #pragma unroll
        for (int di = 0; di < 2; di++)
            #pragma unroll
            for (int dj = 0; dj < 2; dj++) {
                int i = 2 * pi + di, j = 2 * pj + dj;
                float d = D[(size_t)(n * 36 + i * 6 + j) * 100 + k];
                float y = expf(coef[i * 6 + j] * d * d);
                y = (y >= bias) ? y : 0.f;
                out += al[di * 2 + dj] * y;
            }
        out *= 0.25f;
    }
    P2[t] = out;
}

// ---------------- finalize stage2: pool 3 along j with 0.9^(2-j) -> P3 --------
__global__ void f2_k(const float* __restrict__ D, const float* __restrict__ coef,
                     const float* __restrict__ cb, float* __restrict__ P3, int N) {
    int t = blockIdx.x * blockDim.x + threadIdx.x;
    int total = N * 3 * 240;
    if (t >= total) return;
    int k = t % 240, row3 = t / 240;
    float out = 0.f;
    if (k < 225) {
        int n = row3 / 3, i = row3 % 3;
        float c0 = coef[0], bias = cb[1];
        const float al[3] = {0.81f, 0.9f, 1.0f};
        #pragma unroll
        for (int j = 0; j < 3; j++) {
            float d = D[(size_t)(n * 9 + i * 3 + j) * 225 + k];
            float y = expf(c0 * d * d);
            y = (y >= bias) ? y : 0.f;
            out += al[j] * y;
        }
        out *= (1.f / 3.f);
    }
    P3[t] = out;
}

// ---------------- finalize stage3: pool 3 along i with 0.9^(2-i) -> P4 --------
__global__ void f3_k(const float* __restrict__ D, const float* __restrict__ coef,
                     const float* __restrict__ cb, float* __restrict__ P4, int N) {
    int t = blockIdx.x * blockDim.x + threadIdx.x;
    int total = N * 640;
    if (t >= total) return;
    int k = t % 640, n = t / 640;
    float out = 0.f;
    if (k < 625) {
        float c0 = coef[0], bias = cb[2];
        const float al[3] = {0.81f, 0.9f, 1.0f};
        #pragma unroll
        for (int i = 0; i < 3; i++) {
            float d = D[(size_t)(n * 3 + i) * 625 + k];
            float y = expf(c0 * d * d);
            y = (y >= bias) ? y : 0.f;
            out += al[i] * y;
        }
        out *= (1.f / 3.f);
    }
    P4[t] = out;
}

// ---------------- stage4 gauss+crelu fused with FC accumulation ---------------
__global__ void __launch_bounds__(128)
fc_k(const float* __restrict__ D, const float* __restrict__ coef,
     const float* __restrict__ cb, const float* __restrict__ fcw,
     const float* __restrict__ fcb, float* __restrict__ out, int c) {
    __shared__ float y[1232];
    __shared__ float red[128];
    int n = blockIdx.x;
    float c0 = coef[0], bias = cb[3];
    for (int k = threadIdx.x; k < 1225; k += 128) {
        float d = D[(size_t)n * 1225 + k];
        float v = expf(c0 * d * d);
        y[k] = (v >= bias) ? v : 0.f;
    }
    __syncthreads();
    for (int o = 0; o < 10; o++) {
        float p = 0.f;
        const float* wrow = fcw + (size_t)o * 3675 + c * 1225;
        for (int k = threadIdx.x; k < 1225; k += 128) p += y[k] * wrow[k];
        red[threadIdx.x] = p; __syncthreads();
        for (int s = 64; s > 0; s >>= 1) {
            if (threadIdx.x < s) red[threadIdx.x] += red[threadIdx.x + s];
            __syncthreads();
        }
        if (threadIdx.x == 0) {
            float v = red[0];
            if (c == 0) out[n * 10 + o] = fcb[o] + v;
            else        out[n * 10 + o] += v;
        }
        __syncthreads();
    }
}

// ---------------- row-norm |p|^2 for stages 2-4 -------------------------------
__global__ void __launch_bounds__(128)
norm_k(const float* __restrict__ P, float* __restrict__ pn, int Kpad) {
    __shared__ float red[128];
    int r = blockIdx.x;
    const float* row = P + (size_t)r * Kpad;
    float s = 0.f;
    for (int k = threadIdx.x; k < Kpad; k += 128) { float v = row[k]; s += v * v; }
    red[threadIdx.x] = s; __syncthreads();
    for (int st = 64; st > 0; st >>= 1) {
        if (threadIdx.x < st) red[threadIdx.x] += red[threadIdx.x + st];
        __syncthreads();
    }
    if (threadIdx.x == 0) pn[r] = red[0];
}

extern "C" void kernel_launch(void* const* d_in, const int* in_sizes, int n_in,
                              void* d_out, int out_size, void* d_ws, size_t ws_size,
                              hipStream_t stream) {
    const float* x   = (const float*)d_in[0];
    const float* w1  = (const float*)d_in[1];
    const float* w2  = (const float*)d_in[2];
    const float* w3  = (const float*)d_in[3];
    const float* w4  = (const float*)d_in[4];
    const float* fcw = (const float*)d_in[5];
    const float* fcb = (const float*)d_in[6];
    const float* cb  = (const float*)d_in[7];
    float* out = (float*)d_out;
    const int N = in_sizes[0] / (3 * 28 * 28);   // 4096

    float* ws = (float*)d_ws;
    size_t off = 0;
    float* D   = ws + off; off += (size_t)N * 36 * 100;  // dist buffer (max stage1)
    float* Pa  = ws + off; off += (size_t)N * 36 * 32;   // P1 / P3
    float* Pb  = ws + off; off += (size_t)N * 9 * 112;   // P2 / P4
    float* pn  = ws + off; off += (size_t)N * 36;        // row norms
    float* wn  = ws + off; off += 1280;                  // col norms
    float* bsw = ws + off; off += (size_t)640 * 1232;    // swizzled B (max stage4)
    float* sum = ws + off; off += 64;
    float* ssq = ws + off; off += 64;
    float* cof = ws + off; off += 64;

    for (int c = 0; c < 3; c++) {
        // ---- stage 1: K=25->32, O=100->112, R=N*36, 36 window positions ----
        wprep_k<<<2, 64, 0, stream>>>(w1 + (size_t)c * 100 * 25, bsw, wn, 25, 32, 100, 112);
        im2col1_k<<<(N * 36 + 255) / 256, 256, 0, stream>>>(x, c, Pa, pn, N);
        zero_k<<<1, 128, 0, stream>>>(sum, 128);
        gemm_dist_k<4><<<N * 36 / 64, 256, 64 * 32 * 4, stream>>>(Pa, bsw, pn, wn, D, 32, 100, 112);
        reduce_k<<<N * 36, 128, 0, stream>>>(D, sum, ssq, 100, 36);
        coef_k<<<1, 64, 0, stream>>>(sum, ssq, cof, 36, (float)N * 100.f);
        f1_k<<<(N * 9 * 112 + 255) / 256, 256, 0, stream>>>(D, cof, cb, Pb, N);
        norm_k<<<N * 9, 128, 0, stream>>>(Pb, pn, 112);

        // ---- stage 2: K=100->112, O=225->240, R=N*9 ----
        wprep_k<<<4, 64, 0, stream>>>(w2 + (size_t)c * 225 * 100, bsw, wn, 100, 112, 225, 240);
        zero_k<<<1, 128, 0, stream>>>(sum, 128);
        gemm_dist_k<4><<<N * 9 / 64, 256, 64 * 112 * 4, stream>>>(Pb, bsw, pn, wn, D, 112, 225, 240);
        reduce_k<<<N * 9, 128, 0, stream>>>(D, sum, ssq, 225, 1);
        coef_k<<<1, 64, 0, stream>>>(sum, ssq, cof, 1, (float)N * 9 * 225.f);
        f2_k<<<(N * 3 * 240 + 255) / 256, 256, 0, stream>>>(D, cof, cb, Pa, N);
        norm_k<<<N * 3, 128, 0, stream>>>(Pa, pn, 240);

        // ---- stage 3: K=225->240, O=625->640, R=N*3 ----
        wprep_k<<<10, 64, 0, stream>>>(w3 + (size_t)c * 625 * 225, bsw, wn, 225, 240, 625, 640);
        zero_k<<<1, 128, 0, stream>>>(sum, 128);
        gemm_dist_k<2><<<N * 3 / 32, 256, 32 * 240 * 4, stream>>>(Pa, bsw, pn, wn, D, 240, 625, 640);
        reduce_k<<<N * 3, 128, 0, stream>>>(D, sum, ssq, 625, 1);
        coef_k<<<1, 64, 0, stream>>>(sum, ssq, cof, 1, (float)N * 3 * 625.f);
        f3_k<<<(N * 640 + 255) / 256, 256, 0, stream>>>(D, cof, cb, Pb, N);
        norm_k<<<N, 128, 0, stream>>>(Pb, pn, 640);

        // ---- stage 4: K=625->640, O=1225->1232, R=N ----
        wprep_k<<<20, 64, 0, stream>>>(w4 + (size_t)c * 1225 * 625, bsw, wn, 625, 640, 1225, 1232);
        zero_k<<<1, 128, 0, stream>>>(sum, 128);
        gemm_dist_k<1><<<N / 16, 256, 16 * 640 * 4, stream>>>(Pb, bsw, pn, wn, D, 640, 1225, 1232);
        reduce_k<<<N, 128, 0, stream>>>(D, sum, ssq, 1225, 1);
        coef_k<<<1, 64, 0, stream>>>(sum, ssq, cof, 1, (float)N * 1225.f);
        fc_k<<<N, 128, 0, stream>>>(D, cof, cb, fcw, fcb, out, c);
    }
    (void)n_in; (void)out_size; (void)ws_size;
}